// ProbsToSpan_49143015801170
// MI455X (gfx1250) — compile-verified
//
#include <hip/hip_runtime.h>
#include <math.h>

// Problem constants from the reference: B=16, S=4096, fp32 in/out.
#define BATCH 16
#define SLEN  4096
#define NT    256                 // 8 wave32 per workgroup
#define CHUNK (SLEN / NT)         // 16 elements per thread

// ---- CDNA5 async global->LDS path detection -------------------------------
#if defined(__has_builtin)
#  if __has_builtin(__builtin_amdgcn_global_load_async_to_lds_b128)
#    define HAVE_ASYNC_LDS 1
#  endif
#  if __has_builtin(__builtin_amdgcn_s_wait_asynccnt)
#    define HAVE_WAIT_ASYNC 1
#  endif
#endif
#ifndef HAVE_ASYNC_LDS
#  define HAVE_ASYNC_LDS 0
#endif
#ifndef HAVE_WAIT_ASYNC
#  define HAVE_WAIT_ASYNC 0
#endif

// Builtin signature (from clang diagnostic): (v4i AS1*, v4i AS3*, imm, imm)
typedef int v4i __attribute__((ext_vector_type(4)));
typedef __attribute__((address_space(1))) v4i gv4i;   // global (AS1)
typedef __attribute__((address_space(3))) v4i lv4i;   // LDS (AS3)

__device__ __forceinline__ void async_wait_zero() {
#if HAVE_WAIT_ASYNC
  __builtin_amdgcn_s_wait_asynccnt(0);
#else
  asm volatile("s_wait_asynccnt 0" ::: "memory");
#endif
}

// One workgroup per batch row. Computes
//   out[b]        = argmax_s( start[s] * max_{t>=s} end[t] )   (predicted_start)
//   out[BATCH+b]  = argmax_t( end[t]   * max_{s<=t} start[s] ) (predicted_end)
__global__ __launch_bounds__(NT)
void ProbsToSpan_kernel(const float* __restrict__ start_p,
                        const float* __restrict__ end_p,
                        float* __restrict__ out) {
  __shared__ alignas(16) float s_start[SLEN];
  __shared__ alignas(16) float s_end[SLEN];
  __shared__ float pbuf[NT];   // inclusive prefix-max of start chunk-maxes
  __shared__ float sbuf[NT];   // inclusive suffix-max of end chunk-maxes
  __shared__ float rvS[NT]; __shared__ int riS[NT];
  __shared__ float rvE[NT]; __shared__ int riE[NT];

  const int b    = blockIdx.x;
  const int tid  = threadIdx.x;
  const int base = tid * CHUNK;
  const float* gs = start_p + (size_t)b * SLEN;
  const float* ge = end_p   + (size_t)b * SLEN;

  // ---- Stage both rows into LDS (async DMA, bypasses VGPRs) ----------------
#if HAVE_ASYNC_LDS
  // 4 floats per b128 packet; CHUNK/4 = 4 packets per array per thread.
  #pragma unroll
  for (int j = 0; j < CHUNK; j += 4) {
    const int idx = base + j;
    __builtin_amdgcn_global_load_async_to_lds_b128(
        (gv4i*)(void*)(gs + idx), (lv4i*)&s_start[idx], 0, 0);
    __builtin_amdgcn_global_load_async_to_lds_b128(
        (gv4i*)(void*)(ge + idx), (lv4i*)&s_end[idx], 0, 0);
  }
  async_wait_zero();
#else
  #pragma unroll
  for (int j = 0; j < CHUNK; j += 4) {
    const int idx = base + j;
    *(float4*)&s_start[idx] = *(const float4*)(gs + idx);
    *(float4*)&s_end[idx]   = *(const float4*)(ge + idx);
  }
#endif
  __syncthreads();

  // ---- Per-chunk maxes -----------------------------------------------------
  float pmax = -INFINITY, smax = -INFINITY;
  #pragma unroll
  for (int j = 0; j < CHUNK; ++j) {
    pmax = fmaxf(pmax, s_start[base + j]);
    smax = fmaxf(smax, s_end[base + j]);
  }
  pbuf[tid] = pmax;
  sbuf[tid] = smax;
  __syncthreads();

  // ---- Hillis-Steele max-scans over the 256 chunk maxes (8 steps) ----------
  for (int off = 1; off < NT; off <<= 1) {
    const float vp = (tid >= off)      ? pbuf[tid - off] : -INFINITY; // prefix
    const float vs = (tid + off < NT)  ? sbuf[tid + off] : -INFINITY; // suffix
    __syncthreads();
    pbuf[tid] = fmaxf(pbuf[tid], vp);
    sbuf[tid] = fmaxf(sbuf[tid], vs);
    __syncthreads();
  }
  const float excl_p = (tid > 0)      ? pbuf[tid - 1] : -INFINITY;
  const float excl_s = (tid < NT - 1) ? sbuf[tid + 1] : -INFINITY;

  // ---- Forward walk: predicted_end score = end[t] * prefixmax_start[t] -----
  float bestE = -INFINITY; int bestEi = base;
  {
    float run = excl_p;
    #pragma unroll
    for (int j = 0; j < CHUNK; ++j) {
      const int t = base + j;
      run = fmaxf(run, s_start[t]);
      const float sc = s_end[t] * run;
      if (sc > bestE) { bestE = sc; bestEi = t; }   // keep first max index
    }
  }
  // ---- Backward walk: predicted_start score = start[s] * suffixmax_end[s] --
  float bestS = -INFINITY; int bestSi = base + CHUNK - 1;
  {
    float run = excl_s;
    #pragma unroll
    for (int j = CHUNK - 1; j >= 0; --j) {
      const int s = base + j;
      run = fmaxf(run, s_end[s]);
      const float sc = s_start[s] * run;
      if (sc >= bestS) { bestS = sc; bestSi = s; }  // >=: smaller index wins
    }
  }

  rvS[tid] = bestS; riS[tid] = bestSi;
  rvE[tid] = bestE; riE[tid] = bestEi;

  // ---- Block argmax reduction (smaller index wins ties) --------------------
  for (int off = NT / 2; off > 0; off >>= 1) {
    __syncthreads();
    if (tid < off) {
      const int o = tid + off;
      if (rvS[o] > rvS[tid] || (rvS[o] == rvS[tid] && riS[o] < riS[tid])) {
        rvS[tid] = rvS[o]; riS[tid] = riS[o];
      }
      if (rvE[o] > rvE[tid] || (rvE[o] == rvE[tid] && riE[o] < riE[tid])) {
        rvE[tid] = rvE[o]; riE[tid] = riE[o];
      }
    }
  }
  __syncthreads();
  if (tid == 0) {
    out[b]         = (float)riS[0];   // predicted_start
    out[BATCH + b] = (float)riE[0];   // predicted_end
  }
}

extern "C" void kernel_launch(void* const* d_in, const int* in_sizes, int n_in,
                              void* d_out, int out_size, void* d_ws, size_t ws_size,
                              hipStream_t stream) {
  (void)in_sizes; (void)n_in; (void)out_size; (void)d_ws; (void)ws_size;
  const float* start_p = (const float*)d_in[0];
  const float* end_p   = (const float*)d_in[1];
  float* out = (float*)d_out;
  ProbsToSpan_kernel<<<BATCH, NT, 0, stream>>>(start_p, end_p, out);
}